// GraphSAGELinkPrediction_4879082849097
// MI455X (gfx1250) — compile-verified
//
#include <hip/hip_runtime.h>
#include <hip/hip_bf16.h>

typedef __attribute__((ext_vector_type(2))) float v2f;
typedef __attribute__((ext_vector_type(8))) float v8f;

#define MAX_ID_CLAMP 999999
#define DIM 128

// ---------------------------------------------------------------------------
// Gather: out[i,:] = emb[valid(nodes[nb[i]]), :]   (wave per row, float4 lanes)
// ---------------------------------------------------------------------------
__global__ void gather_kernel(const float* __restrict__ emb,
                              const int* __restrict__ nodes,
                              const int* __restrict__ nb,
                              float* __restrict__ out, int n) {
    int row  = blockIdx.x * 8 + (threadIdx.x >> 5);
    int lane = threadIdx.x & 31;
    if (row >= n) return;
    int node = nodes[nb[row]];
    int v = (node >= 0) ? node : MAX_ID_CLAMP;
    const float4* src = (const float4*)(emb + (size_t)v * DIM);
    float4*       dst = (float4*)(out + (size_t)row * DIM);
    dst[lane] = src[lane];                      // 32 lanes x 16B = 512B row
}

// ---------------------------------------------------------------------------
// Fused gather + mean: out[g,:] = (1/S) * sum_s emb[valid(nodes[nb[g*S+s]]),:]
// Avoids materializing the 131 MB hidden[2] tensor.
// ---------------------------------------------------------------------------
__global__ void gather_mean_kernel(const float* __restrict__ emb,
                                   const int* __restrict__ nodes,
                                   const int* __restrict__ nb,
                                   float* __restrict__ out, int n_groups, int S) {
    int g    = blockIdx.x * 8 + (threadIdx.x >> 5);
    int lane = threadIdx.x & 31;
    if (g >= n_groups) return;
    float ax = 0.f, ay = 0.f, az = 0.f, aw = 0.f;
    for (int s = 0; s < S; ++s) {
        int node = nodes[nb[g * S + s]];
        int v = (node >= 0) ? node : MAX_ID_CLAMP;
        float4 x = ((const float4*)(emb + (size_t)v * DIM))[lane];
        ax += x.x; ay += x.y; az += x.z; aw += x.w;
    }
    float inv = 1.0f / (float)S;
    float4 r; r.x = ax * inv; r.y = ay * inv; r.z = az * inv; r.w = aw * inv;
    ((float4*)(out + (size_t)g * DIM))[lane] = r;
}

// ---------------------------------------------------------------------------
// Row-group mean over width-256 rows: out[g,:] = mean_{s<S} in[g*S+s,:]
// ---------------------------------------------------------------------------
__global__ void mean_rows256_kernel(const float* __restrict__ in,
                                    float* __restrict__ out, int n_groups, int S) {
    int g    = blockIdx.x * 8 + (threadIdx.x >> 5);
    int lane = threadIdx.x & 31;
    if (g >= n_groups) return;
    float a0 = 0, a1 = 0, a2 = 0, a3 = 0, b0 = 0, b1 = 0, b2 = 0, b3 = 0;
    for (int s = 0; s < S; ++s) {
        const float4* r = (const float4*)(in + (size_t)(g * S + s) * 256);
        float4 x = r[lane];      a0 += x.x; a1 += x.y; a2 += x.z; a3 += x.w;
        float4 y = r[lane + 32]; b0 += y.x; b1 += y.y; b2 += y.z; b3 += y.w;
    }
    float inv = 1.0f / (float)S;
    float4 r0; r0.x = a0 * inv; r0.y = a1 * inv; r0.z = a2 * inv; r0.w = a3 * inv;
    float4 r1; r1.x = b0 * inv; r1.y = b1 * inv; r1.z = b2 * inv; r1.w = b3 * inv;
    ((float4*)(out + (size_t)g * 256))[lane]      = r0;
    ((float4*)(out + (size_t)g * 256))[lane + 32] = r1;
}

// ---------------------------------------------------------------------------
// FP32 WMMA GEMM:  C[:, col_off:col_off+128] (+= relu) = A[M,K] * W[K,128]
// One wave -> one 16x16 tile via V_WMMA_F32_16X16X4_F32 (K-step 4).
// Block = 256 threads (8 waves) -> 16 rows x 128 cols; W staged via 32KB LDS.
// Requires: M % 16 == 0, K % 64 == 0 (always true here).
// ---------------------------------------------------------------------------
__global__ void gemm_wmma_kernel(const float* __restrict__ A,
                                 const float* __restrict__ W,
                                 float* __restrict__ C,
                                 int M, int K, int lda, int ldc,
                                 int col_off, int do_relu) {
    __shared__ float sB[64 * 128];              // one K-chunk of W (32 KB)
    const int wave = threadIdx.x >> 5;
    const int lane = threadIdx.x & 31;
    const int rowBase = blockIdx.x * 16;
    const int colBase = wave * 16;              // 8 waves cover N=128
    const int hi = lane >> 4;                   // 0: lanes 0-15, 1: lanes 16-31

    v8f acc = {};
    const float* Arow = A + (size_t)(rowBase + (lane & 15)) * lda;

    for (int kc = 0; kc < K; kc += 64) {
        __syncthreads();
        for (int t = threadIdx.x; t < (64 * 128) / 4; t += 256) {
            int idx = t * 4;
            int kr = idx >> 7, cc = idx & 127;
            *(float4*)&sB[idx] = *(const float4*)&W[(size_t)(kc + kr) * 128 + cc];
        }
        __syncthreads();
        for (int k4 = 0; k4 < 64; k4 += 4) {
            // A frag (16x4 f32): lanes 0-15 hold K = k4+{0,1}; lanes 16-31 K = k4+{2,3}
            v2f a;
            a.x = Arow[kc + k4 + 2 * hi + 0];
            a.y = Arow[kc + k4 + 2 * hi + 1];
            // B frag (4x16 f32): rows striped across lane halves per VGPR
            v2f b;
            int bc = colBase + (lane & 15);
            b.x = sB[(k4 + 0 + hi) * 128 + bc];
            b.y = sB[(k4 + 2 + hi) * 128 + bc];
            acc = __builtin_amdgcn_wmma_f32_16x16x4_f32(
                false, a, false, b, (short)0, acc, false, false);
        }
    }

    // C/D 16x16 f32 layout: VGPR r -> (M=r, N=lane) lanes 0-15; (M=8+r) lanes 16-31
    const int cCol  = col_off + colBase + (lane & 15);
    const int rTop  = rowBase + hi * 8;
    #pragma unroll
    for (int r = 0; r < 8; ++r) {
        float v = acc[r];
        if (do_relu) v = fmaxf(v, 0.0f);
        C[(size_t)(rTop + r) * ldc + cCol] = v;
    }
}

// ---------------------------------------------------------------------------
// Row-wise L2 normalize (width 256), wave32 shuffle reduction
// ---------------------------------------------------------------------------
__global__ void l2norm_kernel(float* __restrict__ data, int n) {
    int row  = blockIdx.x * 8 + (threadIdx.x >> 5);
    int lane = threadIdx.x & 31;
    if (row >= n) return;
    float* p = data + (size_t)row * 256;
    float ss = 0.f;
    for (int i = lane; i < 256; i += 32) { float v = p[i]; ss += v * v; }
    for (int off = 16; off > 0; off >>= 1) ss += __shfl_xor(ss, off, 32);
    float scale = rsqrtf(fmaxf(ss, 1e-12f));
    for (int i = lane; i < 256; i += 32) p[i] *= scale;
}

// ---------------------------------------------------------------------------
// logits -> predictions + per-row entropy
// ---------------------------------------------------------------------------
__global__ void dot_pred_kernel(const float* __restrict__ so,
                                const float* __restrict__ dsto,
                                const float* __restrict__ labels,
                                float* __restrict__ pred,
                                float* __restrict__ entropy, int n) {
    int row  = blockIdx.x * 8 + (threadIdx.x >> 5);
    int lane = threadIdx.x & 31;
    if (row >= n) return;
    float d = 0.f;
    for (int i = lane; i < 256; i += 32)
        d += so[(size_t)row * 256 + i] * dsto[(size_t)row * 256 + i];
    for (int off = 16; off > 0; off >>= 1) d += __shfl_xor(d, off, 32);
    if (lane == 0) {
        float logit = d;
        pred[row] = 1.0f / (1.0f + expf(-logit));
        float z = labels[row];
        entropy[row] = fmaxf(logit, 0.0f) - logit * z + log1pf(expf(-fabsf(logit)));
    }
}

__global__ void loss_reduce_kernel(const float* __restrict__ entropy,
                                   float* __restrict__ loss_out, int n) {
    __shared__ float sm[256];
    float s = 0.f;
    for (int i = threadIdx.x; i < n; i += 256) s += entropy[i];
    sm[threadIdx.x] = s;
    __syncthreads();
    for (int off = 128; off > 0; off >>= 1) {
        if ((int)threadIdx.x < off) sm[threadIdx.x] += sm[threadIdx.x + off];
        __syncthreads();
    }
    if (threadIdx.x == 0) *loss_out = sm[0] / (float)n;
}

// ---------------------------------------------------------------------------
// Host orchestration
// ---------------------------------------------------------------------------
extern "C" void kernel_launch(void* const* d_in, const int* in_sizes, int n_in,
                              void* d_out, int out_size, void* d_ws, size_t ws_size,
                              hipStream_t stream) {
    const int B  = 1024;
    const int H1 = 10240;

    const int*   nodes[2] = {(const int*)d_in[0], (const int*)d_in[1]};
    const float* labels   = (const float*)d_in[2];
    const int*   nb0[2]   = {(const int*)d_in[3], (const int*)d_in[4]};
    const int*   nb1[2]   = {(const int*)d_in[5], (const int*)d_in[6]};
    const int*   nb2[2]   = {(const int*)d_in[7], (const int*)d_in[8]};
    const float* emb      = (const float*)d_in[9];
    // per side: w_self_0, w_neigh_0, w_self_1, w_neigh_1
    const float* w[2][4] = {
        {(const float*)d_in[10], (const float*)d_in[11],
         (const float*)d_in[12], (const float*)d_in[13]},
        {(const float*)d_in[14], (const float*)d_in[15],
         (const float*)d_in[16], (const float*)d_in[17]}};

    float* ws = (float*)d_ws;

    // per-side workspace layout (floats)
    const size_t O_H0  = 0;                       // [1024,128]
    const size_t O_H1  = O_H0  + (size_t)B  * 128;  // [10240,128]
    const size_t O_M0  = O_H1  + (size_t)H1 * 128;  // [1024,128]
    const size_t O_M1  = O_M0  + (size_t)B  * 128;  // [10240,128]
    const size_t O_Y0  = O_M1  + (size_t)H1 * 128;  // [1024,256]
    const size_t O_Y1  = O_Y0  + (size_t)B  * 256;  // [10240,256]
    const size_t O_MM1 = O_Y1  + (size_t)H1 * 256;  // [1024,256]
    const size_t O_OUT = O_MM1 + (size_t)B  * 256;  // [1024,256]
    const size_t SIDE  = O_OUT + (size_t)B  * 256;  // floats per side
    float* entropy = ws + 2 * SIDE;                 // [1024]

    dim3 blk(256);
    auto gemm = [&](const float* A, const float* Wm, float* C,
                    int M, int K, int lda, int col_off, int relu) {
        gemm_wmma_kernel<<<dim3(M / 16), blk, 0, stream>>>(
            A, Wm, C, M, K, lda, 256, col_off, relu);
    };

    for (int s = 0; s < 2; ++s) {
        float* base = ws + (size_t)s * SIDE;
        float* h0  = base + O_H0;
        float* h1  = base + O_H1;
        float* m0  = base + O_M0;
        float* m1  = base + O_M1;
        float* y0  = base + O_Y0;
        float* y1  = base + O_Y1;
        float* mm1 = base + O_MM1;
        float* out = base + O_OUT;

        // gathers + fused neighbor means (dominant HBM traffic)
        gather_kernel<<<dim3(B / 8),  blk, 0, stream>>>(emb, nodes[s], nb0[s], h0, B);
        gather_kernel<<<dim3(H1 / 8), blk, 0, stream>>>(emb, nodes[s], nb1[s], h1, H1);
        gather_mean_kernel<<<dim3(B / 8),  blk, 0, stream>>>(emb, nodes[s], nb1[s], m0, B, 10);
        gather_mean_kernel<<<dim3(H1 / 8), blk, 0, stream>>>(emb, nodes[s], nb2[s], m1, H1, 25);

        // layer 0: concat(self@W, mean@W) with ReLU
        gemm(h1, w[s][0], y1, H1, 128, 128,   0, 1);
        gemm(m1, w[s][1], y1, H1, 128, 128, 128, 1);
        gemm(h0, w[s][0], y0, B,  128, 128,   0, 1);
        gemm(m0, w[s][1], y0, B,  128, 128, 128, 1);

        // layer 1: mean over 10 samples of y1, then concat GEMMs (K=256), no act
        mean_rows256_kernel<<<dim3(B / 8), blk, 0, stream>>>(y1, mm1, B, 10);
        gemm(y0,  w[s][2], out, B, 256, 256,   0, 0);
        gemm(mm1, w[s][3], out, B, 256, 256, 128, 0);

        l2norm_kernel<<<dim3(B / 8), blk, 0, stream>>>(out, B);
    }

    float* src_out = ws + O_OUT;
    float* dst_out = ws + SIDE + O_OUT;
    float* pred = (float*)d_out;          // [1024] predictions
    float* loss = pred + B;               // [1] loss

    dot_pred_kernel<<<dim3(B / 8), blk, 0, stream>>>(src_out, dst_out, labels,
                                                     pred, entropy, B);
    loss_reduce_kernel<<<dim3(1), blk, 0, stream>>>(entropy, loss, B);
}